// RGPRGNN_14766097564159
// MI455X (gfx1250) — compile-verified
//
#include <hip/hip_runtime.h>
#include <hip/hip_bf16.h>

// Problem constants (match reference)
#define NN   50000
#define EE   600000
#define INC  256
#define HIDC 128
#define OUTC 2
#define RR   7
#define BBASES 8
#define LLAYERS 3
#define NT_M 3125   // 50000/16 tiles along node dim
#define NT_F 8      // 128/16 tiles along feature dim

typedef float v2f __attribute__((ext_vector_type(2)));
typedef float v8f __attribute__((ext_vector_type(8)));

// ---------------------------------------------------------------------------
// Graph preprocessing: per-(dst,relation) edge counts -> inverse counts
// ---------------------------------------------------------------------------
__global__ __launch_bounds__(256) void count_edges(const int* __restrict__ dst,
                                                   const int* __restrict__ et,
                                                   int* __restrict__ cnt) {
    int e = blockIdx.x * blockDim.x + threadIdx.x;
    if (e < EE) atomicAdd(&cnt[dst[e] * RR + et[e]], 1);
}

__global__ __launch_bounds__(256) void make_inv(const int* __restrict__ cnt,
                                                float* __restrict__ inv) {
    int i = blockIdx.x * blockDim.x + threadIdx.x;
    if (i < NN * RR) {
        int c = cnt[i];
        inv[i] = 1.0f / (float)(c > 1 ? c : 1);
    }
}

// ---------------------------------------------------------------------------
// W[l,r,:,:] = sum_b comps[l,r,b] * bases[l,b,:,:]   (tiny: 344K outputs)
// ---------------------------------------------------------------------------
__global__ __launch_bounds__(256) void build_w(const float* __restrict__ comps,
                                               const float* __restrict__ bases,
                                               float* __restrict__ Wall) {
    int t = blockIdx.x * blockDim.x + threadIdx.x;
    const int total = LLAYERS * RR * HIDC * HIDC;
    if (t >= total) return;
    int f  = t & (HIDC - 1);
    int d  = (t >> 7) & (HIDC - 1);
    int rl = t >> 14;           // l*RR + r
    int r  = rl % RR;
    int l  = rl / RR;
    float acc = 0.f;
#pragma unroll
    for (int b = 0; b < BBASES; ++b)
        acc += comps[(l * RR + r) * BBASES + b] *
               bases[(((size_t)l * BBASES + b) * HIDC + d) * HIDC + f];
    Wall[t] = acc;
}

// ---------------------------------------------------------------------------
// Edge scatter: one wave per edge; lane moves a float4 (32 lanes * 4 = 128)
// s[(dst*R+et), :] += cur[src, :]
// ---------------------------------------------------------------------------
__global__ __launch_bounds__(256) void scatter_add(const float* __restrict__ cur,
                                                   const int* __restrict__ src,
                                                   const int* __restrict__ dst,
                                                   const int* __restrict__ et,
                                                   float* __restrict__ s) {
    int wave = (blockIdx.x * blockDim.x + threadIdx.x) >> 5;
    int lane = threadIdx.x & 31;
    if (wave >= EE) return;
    int sn = src[wave];
    int dn = dst[wave];
    int r  = et[wave];
    const float4 v = ((const float4*)(cur + (size_t)sn * HIDC))[lane];
    float* o = s + ((size_t)dn * RR + r) * HIDC + lane * 4;
    atomicAdd(o + 0, v.x);
    atomicAdd(o + 1, v.y);
    atomicAdd(o + 2, v.z);
    atomicAdd(o + 3, v.w);
}

// ---------------------------------------------------------------------------
// lin1 + GPR init: hidden = cur0 = temp[0] * (x @ lin1_w + lin1_b)
// One wave = 16 nodes x 128 features (8 WMMA f32 16x16x4 tiles, K=256)
// ---------------------------------------------------------------------------
__global__ __launch_bounds__(128) void lin1_wmma(const float* __restrict__ x,
                                                 const float* __restrict__ w,
                                                 const float* __restrict__ b,
                                                 const float* __restrict__ temp,
                                                 float* __restrict__ hidden,
                                                 float* __restrict__ cur) {
    int wid  = threadIdx.x >> 5;
    int lane = threadIdx.x & 31;
    int tile_m = blockIdx.x * 4 + wid;
    if (tile_m >= NT_M) return;
    int  lo = lane & 15;
    bool hi = lane >= 16;

    v8f acc[NT_F];
#pragma unroll
    for (int j = 0; j < NT_F; ++j) {
        float bv = b[j * 16 + lo];
#pragma unroll
        for (int v = 0; v < 8; ++v) acc[j][v] = bv;
    }

    const float* A = x + (size_t)tile_m * 16 * INC + (size_t)lo * INC;
    for (int k = 0; k < INC; k += 4) {
        int ka = k + (hi ? 2 : 0);
        float2 av = *(const float2*)(A + ka);
        v2f a; a.x = av.x; a.y = av.y;
#pragma unroll
        for (int j = 0; j < NT_F; ++j) {
            v2f bb;
            bb.x = w[(size_t)ka * HIDC + j * 16 + lo];
            bb.y = w[(size_t)(ka + 1) * HIDC + j * 16 + lo];
            acc[j] = __builtin_amdgcn_wmma_f32_16x16x4_f32(
                false, a, false, bb, (short)0, acc[j], false, false);
        }
    }

    float t0 = temp[0];
    int mbase = tile_m * 16 + (hi ? 8 : 0);
#pragma unroll
    for (int j = 0; j < NT_F; ++j)
#pragma unroll
        for (int v = 0; v < 8; ++v) {
            float val = acc[j][v] * t0;
            size_t idx = (size_t)(mbase + v) * HIDC + j * 16 + lo;
            hidden[idx] = val;
            cur[idx]    = val;
        }
}

// ---------------------------------------------------------------------------
// Fused RGCN layer + GPR accumulation:
//   out = cur_in @ root + bias + (s * inv) @ Wstack      (K = 128, then 896)
//   if relu: out = max(out,0);  cur_out = out;  hidden += temp[ti] * out
// inv scaling folded into the A-operand load, hoisted per relation block.
// ---------------------------------------------------------------------------
__global__ __launch_bounds__(128) void rgcn_wmma(const float* __restrict__ cur_in,
                                                 const float* __restrict__ s,
                                                 const float* __restrict__ inv,
                                                 const float* __restrict__ Wl,
                                                 const float* __restrict__ root,
                                                 const float* __restrict__ bias,
                                                 const float* __restrict__ temp,
                                                 int tempidx, int dorelu,
                                                 float* __restrict__ cur_out,
                                                 float* __restrict__ hidden) {
    int wid  = threadIdx.x >> 5;
    int lane = threadIdx.x & 31;
    int tile_m = blockIdx.x * 4 + wid;
    if (tile_m >= NT_M) return;
    int  lo = lane & 15;
    bool hi = lane >= 16;

    v8f acc[NT_F];
#pragma unroll
    for (int j = 0; j < NT_F; ++j) {
        float bv = bias[j * 16 + lo];
#pragma unroll
        for (int v = 0; v < 8; ++v) acc[j][v] = bv;
    }

    // --- part 1: cur_in @ root, K = 128 ---
    const float* A1 = cur_in + (size_t)tile_m * 16 * HIDC + (size_t)lo * HIDC;
    for (int k = 0; k < HIDC; k += 4) {
        int ka = k + (hi ? 2 : 0);
        float2 av = *(const float2*)(A1 + ka);
        v2f a; a.x = av.x; a.y = av.y;
#pragma unroll
        for (int j = 0; j < NT_F; ++j) {
            v2f bb;
            bb.x = root[(size_t)ka * HIDC + j * 16 + lo];
            bb.y = root[(size_t)(ka + 1) * HIDC + j * 16 + lo];
            acc[j] = __builtin_amdgcn_wmma_f32_16x16x4_f32(
                false, a, false, bb, (short)0, acc[j], false, false);
        }
    }

    // --- part 2: (s * inv) @ Wstack, K = R*128 = 896 ---
    const float* A2   = s   + (size_t)tile_m * 16 * (RR * HIDC) + (size_t)lo * (RR * HIDC);
    const float* invp = inv + (size_t)(tile_m * 16 + lo) * RR;   // this lane's node
    for (int r = 0; r < RR; ++r) {
        float sc = invp[r];
        const float* A2r = A2 + r * HIDC;
        const float* Wr  = Wl + (size_t)r * HIDC * HIDC;
        for (int k = 0; k < HIDC; k += 4) {
            int ka = k + (hi ? 2 : 0);
            float2 av = *(const float2*)(A2r + ka);
            v2f a; a.x = av.x * sc; a.y = av.y * sc;
#pragma unroll
            for (int j = 0; j < NT_F; ++j) {
                v2f bb;
                bb.x = Wr[(size_t)ka * HIDC + j * 16 + lo];
                bb.y = Wr[(size_t)(ka + 1) * HIDC + j * 16 + lo];
                acc[j] = __builtin_amdgcn_wmma_f32_16x16x4_f32(
                    false, a, false, bb, (short)0, acc[j], false, false);
            }
        }
    }

    float tl = temp[tempidx];
    int mbase = tile_m * 16 + (hi ? 8 : 0);
#pragma unroll
    for (int j = 0; j < NT_F; ++j)
#pragma unroll
        for (int v = 0; v < 8; ++v) {
            float val = acc[j][v];
            if (dorelu) val = fmaxf(val, 0.0f);
            size_t idx = (size_t)(mbase + v) * HIDC + j * 16 + lo;
            cur_out[idx] = val;
            hidden[idx] += tl * val;
        }
}

// ---------------------------------------------------------------------------
// out = hidden @ lin2_w + lin2_b   (F = 2; one wave per node, shuffle reduce)
// ---------------------------------------------------------------------------
__global__ __launch_bounds__(256) void lin2_kernel(const float* __restrict__ hidden,
                                                   const float* __restrict__ w,
                                                   const float* __restrict__ b,
                                                   float* __restrict__ out) {
    int wid  = threadIdx.x >> 5;
    int lane = threadIdx.x & 31;
    int node = blockIdx.x * 8 + wid;
    if (node >= NN) return;
    const float* h = hidden + (size_t)node * HIDC;
    float a0 = 0.f, a1 = 0.f;
    for (int k = lane; k < HIDC; k += 32) {
        float hv = h[k];
        a0 += hv * w[k * 2 + 0];
        a1 += hv * w[k * 2 + 1];
    }
    for (int off = 16; off > 0; off >>= 1) {
        a0 += __shfl_down(a0, off);
        a1 += __shfl_down(a1, off);
    }
    if (lane == 0) {
        out[(size_t)node * 2 + 0] = a0 + b[0];
        out[(size_t)node * 2 + 1] = a1 + b[1];
    }
}

// ---------------------------------------------------------------------------
extern "C" void kernel_launch(void* const* d_in, const int* in_sizes, int n_in,
                              void* d_out, int out_size, void* d_ws, size_t ws_size,
                              hipStream_t stream) {
    const float* x      = (const float*)d_in[0];
    const int*   eidx   = (const int*)d_in[1];
    const int*   srcp   = eidx;
    const int*   dstp   = eidx + EE;
    const int*   et     = (const int*)d_in[2];
    const float* temp   = (const float*)d_in[3];
    const float* lin1_w = (const float*)d_in[4];
    const float* lin1_b = (const float*)d_in[5];
    const float* lin2_w = (const float*)d_in[6];
    const float* lin2_b = (const float*)d_in[7];
    const float* comps  = (const float*)d_in[8];
    const float* bases  = (const float*)d_in[9];
    const float* roots  = (const float*)d_in[10];
    const float* cbias  = (const float*)d_in[11];
    float* out = (float*)d_out;

    // Workspace carving (256B aligned)
    char* ws = (char*)d_ws;
    size_t off = 0;
    auto take = [&](size_t nbytes) -> void* {
        void* p = ws + off;
        off = (off + nbytes + 255) & ~(size_t)255;
        return p;
    };
    float* s      = (float*)take((size_t)NN * RR * HIDC * sizeof(float)); // 179.2 MB
    int*   cnt    = (int*)  take((size_t)NN * RR * sizeof(int));
    float* inv    = (float*)take((size_t)NN * RR * sizeof(float));
    float* Wall   = (float*)take((size_t)LLAYERS * RR * HIDC * HIDC * sizeof(float));
    float* curA   = (float*)take((size_t)NN * HIDC * sizeof(float));
    float* curB   = (float*)take((size_t)NN * HIDC * sizeof(float));
    float* hidden = (float*)take((size_t)NN * HIDC * sizeof(float));
    (void)ws_size; (void)in_sizes; (void)n_in; (void)out_size;

    // Graph-invariant preprocessing (counts reused by all layers)
    hipMemsetAsync(cnt, 0, (size_t)NN * RR * sizeof(int), stream);
    count_edges<<<(EE + 255) / 256, 256, 0, stream>>>(dstp, et, cnt);
    make_inv<<<(NN * RR + 255) / 256, 256, 0, stream>>>(cnt, inv);
    build_w<<<(LLAYERS * RR * HIDC * HIDC + 255) / 256, 256, 0, stream>>>(comps, bases, Wall);

    // hidden = cur0 = temp[0] * (x @ lin1_w + b)
    lin1_wmma<<<dim3((NT_M + 3) / 4), 128, 0, stream>>>(x, lin1_w, lin1_b, temp, hidden, curA);

    float* cin  = curA;
    float* cout = curB;
    for (int l = 0; l < LLAYERS; ++l) {
        hipMemsetAsync(s, 0, (size_t)NN * RR * HIDC * sizeof(float), stream);
        scatter_add<<<(EE + 7) / 8, 256, 0, stream>>>(cin, srcp, dstp, et, s);
        rgcn_wmma<<<dim3((NT_M + 3) / 4), 128, 0, stream>>>(
            cin, s, inv,
            Wall  + (size_t)l * RR * HIDC * HIDC,
            roots + (size_t)l * HIDC * HIDC,
            cbias + (size_t)l * HIDC,
            temp, l + 1, (l < LLAYERS - 1) ? 1 : 0,
            cout, hidden);
        float* t = cin; cin = cout; cout = t;
    }

    lin2_kernel<<<(NN + 7) / 8, 256, 0, stream>>>(hidden, lin2_w, lin2_b, out);
}